// SinkhornRouter_2302102471507
// MI455X (gfx1250) — compile-verified
//
#include <hip/hip_runtime.h>
#include <hip/hip_bf16.h>
#include <math.h>

typedef float v2f __attribute__((ext_vector_type(2)));
typedef float v4f __attribute__((ext_vector_type(4)));
typedef float v8f __attribute__((ext_vector_type(8)));
typedef int   v4i __attribute__((ext_vector_type(4)));

#define TOKENS 8192
#define HS     4096
#define NE     64
#define KT     128            // K-tile staged in LDS
#define WSTRIDE 132           // padded row stride (floats): bank = 4*col mod 64 -> conflict-free

#if __has_builtin(__builtin_amdgcn_global_load_async_to_lds_b128) && \
    __has_builtin(__builtin_amdgcn_s_wait_asynccnt)
#define USE_ASYNC_LDS 1
#endif

typedef __attribute__((address_space(1))) v4i as1_v4i;   // global (printed as __device__)
typedef __attribute__((address_space(3))) v4i as3_v4i;   // LDS (printed as __shared__)

// ---------------------------------------------------------------------------
// Kernel 1: logits = x @ W^T  (M=8192, N=64, K=4096, fp32)  +  cost = exp(logits)
// HBM floor: x is 134 MB -> ~5.8us at 23.3 TB/s; read exactly once (one wave
// owns 16 token rows x all 64 experts, 4x V_WMMA_F32_16X16X4_F32 accumulators).
// W K-tiles (64x128 = 32 KB) are staged into LDS once per block via
// GLOBAL_LOAD_ASYNC_TO_LDS_B128 (ASYNCcnt) and fed to WMMA via ds_load_b64,
// cutting W cache traffic 8x (512 MB -> 64 MB).
// ---------------------------------------------------------------------------
__global__ __launch_bounds__(256) void gemm_logits_kernel(
    const float* __restrict__ x, const float* __restrict__ W,
    float* __restrict__ logits, float* __restrict__ cost)
{
    __shared__ float sW[NE * WSTRIDE];          // 33.8 KB

    const int tid  = threadIdx.x;
    const int wave = tid >> 5;                  // 0..7
    const int lane = tid & 31;
    const int lrow = lane & 15;
    const int khalf = (lane >> 4) << 1;         // 0 or 2 (K sub-offset per ISA A/B layout)
    const int rowBase = (blockIdx.x * 8 + wave) * 16;

    // A: lanes 0-15 load x[row][k..k+1], lanes 16-31 load x[row][k+2..k+3]
    const float* aptr = x + (size_t)(rowBase + lrow) * HS + khalf;

    // LDS base offsets (floats) for the 4 B fragments of this lane
    const int b0off = ( 0 + lrow) * WSTRIDE + khalf;
    const int b1off = (16 + lrow) * WSTRIDE + khalf;
    const int b2off = (32 + lrow) * WSTRIDE + khalf;
    const int b3off = (48 + lrow) * WSTRIDE + khalf;

    v8f acc0, acc1, acc2, acc3;
    for (int r = 0; r < 8; ++r) { acc0[r] = 0.f; acc1[r] = 0.f; acc2[r] = 0.f; acc3[r] = 0.f; }

    for (int k0 = 0; k0 < HS; k0 += KT) {
        __syncthreads();                         // previous tile fully consumed

        // Stage W[0:64][k0:k0+128] -> sW. 64 rows x 32 chunks of 16B = 2048
        // chunks; 8 chunks per thread.
        #pragma unroll
        for (int q = 0; q < (NE * (KT / 4)) / 256; ++q) {
            const int c   = tid + q * 256;
            const int row = c >> 5;              // c / 32
            const int cc  = c & 31;              // 16B chunk within row
            const float* gsrc = W + (size_t)row * HS + k0 + cc * 4;
            float*       ldst = &sW[row * WSTRIDE + cc * 4];
#ifdef USE_ASYNC_LDS
            __builtin_amdgcn_global_load_async_to_lds_b128(
                (as1_v4i*)(uintptr_t)gsrc,
                (as3_v4i*)(unsigned int)(uintptr_t)ldst, 0, 0);
#else
            *(v4f*)ldst = *(const v4f*)gsrc;
#endif
        }
#ifdef USE_ASYNC_LDS
        __builtin_amdgcn_s_wait_asynccnt(0);
#endif
        __syncthreads();

        // Consume the tile: 32 k-steps of 4
        for (int kk = 0; kk < KT; kk += 4) {
            v2f a  = *(const v2f*)aptr;  aptr += 4;
            v2f b0 = *(const v2f*)&sW[b0off + kk];
            v2f b1 = *(const v2f*)&sW[b1off + kk];
            v2f b2 = *(const v2f*)&sW[b2off + kk];
            v2f b3 = *(const v2f*)&sW[b3off + kk];
            acc0 = __builtin_amdgcn_wmma_f32_16x16x4_f32(false, a, false, b0, (short)0, acc0, false, false);
            acc1 = __builtin_amdgcn_wmma_f32_16x16x4_f32(false, a, false, b1, (short)0, acc1, false, false);
            acc2 = __builtin_amdgcn_wmma_f32_16x16x4_f32(false, a, false, b2, (short)0, acc2, false, false);
            acc3 = __builtin_amdgcn_wmma_f32_16x16x4_f32(false, a, false, b3, (short)0, acc3, false, false);
        }
    }

    // C layout: lane<16 -> N=lane, M=r ; lane>=16 -> N=lane-16, M=8+r
    const int moff = (lane >> 4) << 3;          // 0 or 8
    for (int r = 0; r < 8; ++r) {
        const int row = rowBase + r + moff;
        {
            float v = acc0[r]; size_t o = (size_t)row * NE + ( 0 + lrow);
            logits[o] = v; cost[o] = expf(v);
        }
        {
            float v = acc1[r]; size_t o = (size_t)row * NE + (16 + lrow);
            logits[o] = v; cost[o] = expf(v);
        }
        {
            float v = acc2[r]; size_t o = (size_t)row * NE + (32 + lrow);
            logits[o] = v; cost[o] = expf(v);
        }
        {
            float v = acc3[r]; size_t o = (size_t)row * NE + (48 + lrow);
            logits[o] = v; cost[o] = expf(v);
        }
    }
}

// ---------------------------------------------------------------------------
// Kernel 2: Sinkhorn iteration, entirely on-device (graph capture forbids host
// sync, so the data-dependent while-loop must live in one kernel). Single
// 1024-thread workgroup; cost matrix (2 MB) is L2-resident; d0 (32 KB) in LDS.
// Outputs only d1 (64 floats): top-k of d1*cost*d0 == top-k of d1*cost per row
// since d0[i] > 0 is a row-constant scale.
// ---------------------------------------------------------------------------
__global__ __launch_bounds__(1024) void sinkhorn_kernel(
    const float* __restrict__ cost, float* __restrict__ d1_out)
{
    __shared__ float s_d0[TOKENS];
    __shared__ float s_d1[NE];
    __shared__ float s_d1n[NE];
    __shared__ float s_part[16 * NE];
    __shared__ float s_errbuf[NE];
    __shared__ float s_err;

    const int tid = threadIdx.x;
    const float EPSv = 1e-8f, TOLv = 1e-4f;
    const float inv_n0 = 1.0f / (float)TOKENS;
    const float inv_n1 = 1.0f / (float)NE;

    if (tid < NE) s_d1[tid] = 1.0f;
    if (tid == 0) s_err = 1e9f;
    __syncthreads();

    for (int it = 0; it < 10000; ++it) {
        if (s_err <= TOLv) break;   // uniform: s_err written pre-barrier by thread 0

        // Row pass: d0[i] = (1/n0) / (sum_j d1[j]*cost[i,j] + eps)
        for (int rr = 0; rr < TOKENS / 1024; ++rr) {
            const int i = tid + rr * 1024;
            const float* crow = cost + (size_t)i * NE;
            float s = 0.f;
            #pragma unroll 8
            for (int j = 0; j < NE; ++j) s += s_d1[j] * crow[j];
            s_d0[i] = inv_n0 / (s + EPSv);
        }
        __syncthreads();

        // Column pass: thread (g, j) sums 512 rows of d0[i]*cost[i,j]
        {
            const int j = tid & 63;
            const int g = tid >> 6;             // 0..15
            const int i0 = g * (TOKENS / 16);
            float s = 0.f;
            for (int i = i0; i < i0 + TOKENS / 16; ++i)
                s += s_d0[i] * cost[(size_t)i * NE + j];
            s_part[g * NE + j] = s;
        }
        __syncthreads();

        if (tid < NE) {
            float c = 0.f;
            #pragma unroll
            for (int g2 = 0; g2 < 16; ++g2) c += s_part[g2 * NE + tid];
            const float dn = inv_n1 / (c + EPSv);
            s_d1n[tid] = dn;
            s_errbuf[tid] = fabsf(s_d1[tid] - dn);
        }
        __syncthreads();
        if (tid == 0) {
            float e = 0.f;
            #pragma unroll
            for (int q = 0; q < NE; ++q) e += s_errbuf[q];
            s_err = e * inv_n1;
        }
        __syncthreads();
        if (tid < NE) s_d1[tid] = s_d1n[tid];
        __syncthreads();
    }

    if (tid < NE) d1_out[tid] = s_d1[tid];
}

// ---------------------------------------------------------------------------
// Kernel 3: per-token top-2 of d1[j]*exp(logit[j]) + fp32 softmax gather.
// One thread per token; 2 MB of logits read from L2. Ties resolve to the
// lower index first (strict > scan), matching jax.lax.top_k.
// ---------------------------------------------------------------------------
__global__ __launch_bounds__(256) void topk_softmax_kernel(
    const float* __restrict__ logits, const float* __restrict__ d1,
    float* __restrict__ out)
{
    __shared__ float s_d1[NE];
    if (threadIdx.x < NE) s_d1[threadIdx.x] = d1[threadIdx.x];
    __syncthreads();

    const int tok = blockIdx.x * blockDim.x + threadIdx.x;
    if (tok >= TOKENS) return;

    const float* row = logits + (size_t)tok * NE;
    float l[NE];
    float m = -INFINITY;
    #pragma unroll 8
    for (int j = 0; j < NE; ++j) { l[j] = row[j]; m = fmaxf(m, l[j]); }

    float se = 0.f;
    #pragma unroll 8
    for (int j = 0; j < NE; ++j) se += expf(l[j] - m);

    float b1 = -INFINITY, b2 = -INFINITY;
    int i1 = 0, i2 = 0;
    for (int j = 0; j < NE; ++j) {
        const float v = s_d1[j] * expf(l[j]);   // == d1[j]*cost[tok,j]
        if (v > b1)      { b2 = b1; i2 = i1; b1 = v; i1 = j; }
        else if (v > b2) { b2 = v; i2 = j; }
    }

    const float inv_se = 1.0f / se;
    out[(size_t)tok * 2 + 0] = expf(l[i1] - m) * inv_se;
    out[(size_t)tok * 2 + 1] = expf(l[i2] - m) * inv_se;
    out[(size_t)TOKENS * 2 + (size_t)tok * 2 + 0] = (float)i1;
    out[(size_t)TOKENS * 2 + (size_t)tok * 2 + 1] = (float)i2;
}

// ---------------------------------------------------------------------------
extern "C" void kernel_launch(void* const* d_in, const int* in_sizes, int n_in,
                              void* d_out, int out_size, void* d_ws, size_t ws_size,
                              hipStream_t stream) {
    const float* x = (const float*)d_in[0];   // (2048, 4, 4096) fp32 -> 8192 x 4096
    const float* W = (const float*)d_in[1];   // (64, 4096) fp32

    float* logits = (float*)d_ws;                     // 8192*64 = 2 MB
    float* cost   = logits + (size_t)TOKENS * NE;     // 2 MB
    float* d1     = cost   + (size_t)TOKENS * NE;     // 64 floats

    gemm_logits_kernel<<<TOKENS / 128, 256, 0, stream>>>(x, W, logits, cost);
    sinkhorn_kernel<<<1, 1024, 0, stream>>>(cost, d1);
    topk_softmax_kernel<<<TOKENS / 256, 256, 0, stream>>>(logits, d1, (float*)d_out);
}